// chamfer_distance_32650341384461
// MI455X (gfx1250) — compile-verified
//
#include <hip/hip_runtime.h>
#include <hip/hip_bf16.h>

typedef __attribute__((ext_vector_type(2))) float v2f;
typedef __attribute__((ext_vector_type(8))) float v8f;

#define BDIM 4
#define SDIM 4
#define NPTS 4096

// Zero the 2*BDIM*SDIM accumulators in workspace.
__global__ void chamfer_zero_kernel(float* __restrict__ ws) {
    int t = threadIdx.x;
    if (t < 2 * BDIM * SDIM) ws[t] = 0.0f;
}

// For each (b,s), for each 16-row tile of P, find min squared distance to all
// M points of Q per row, sum the 16 row-mins, atomically accumulate per (b,s).
// One wave32 per 16-row tile.
//
// The entire distance formula is computed by a single WMMA per 16x16 tile:
//   A row  = (-2px, -2py, -2pz, 1)      (K=4, loop-invariant)
//   B col  = ( qx,   qy,   qz,  y^2)
//   C[r]   = x^2[row]                   (C/D layout: VGPR r -> row r+8*hi)
//   D      = x^2 + y^2 - 2 x.q  =  squared distance
//
// The running min is kept as signed-int compares on the float bit patterns:
// valid for non-NaN data because (a) for values >= 0 int order == float
// order, and (b) any negative value (fp cancellation) wins the int-min and
// is then clamped to 0 by the final max(.,0) — identical to the reference's
// max(min(d),0) since max(.,0) is monotonic and commutes with min.
__global__ void __launch_bounds__(256)
chamfer_nn_kernel(const float* __restrict__ P, const float* __restrict__ Q,
                  float* __restrict__ accum) {
    const int lane = threadIdx.x & 31;
    const int wave = (blockIdx.x * (blockDim.x >> 5)) + (threadIdx.x >> 5);
    const int tilesPerBS = NPTS / 16;            // 256
    const int bs    = wave / tilesPerBS;         // 0..15
    const int iTile = wave % tilesPerBS;

    const int hi = lane >> 4;                    // 0: K=0,1  1: K=2,3
    const int lo = lane & 15;

    const float* Pbs = P + (size_t)bs * NPTS * 3;
    const float* Qbs = Q + (size_t)bs * NPTS * 3;

    // --- A fragment: 16x4 f32, rows = query points, K = (-2x,-2y,-2z,1) ---
    const int row = iTile * 16 + lo;
    const float px = Pbs[row * 3 + 0];
    const float py = Pbs[row * 3 + 1];
    const float pz = Pbs[row * 3 + 2];
    const float x2self = px * px + py * py + pz * pz;

    v2f a;
    a.x = hi ? (-2.0f * pz) : (-2.0f * px);
    a.y = hi ? 1.0f         : (-2.0f * py);

    // C fragment: x^2 for the 8 rows this lane's C/D VGPRs map to (r + 8*hi).
    v8f x2v;
#pragma unroll
    for (int r = 0; r < 8; ++r)
        x2v[r] = __shfl(x2self, r + 8 * hi, 32);

    int rmin[8];
#pragma unroll
    for (int r = 0; r < 8; ++r) rmin[r] = 0x7f7fffff;   // FLT_MAX bits

    for (int jt = 0; jt < tilesPerBS; ++jt) {
        const int col = jt * 16 + lo;
        const float qx = Qbs[col * 3 + 0];
        const float qy = Qbs[col * 3 + 1];
        const float qz = Qbs[col * 3 + 2];
        const float y2 = qx * qx + qy * qy + qz * qz;

        // --- B fragment: 4x16 f32, cols = target points, K = (x,y,z,y^2) ---
        v2f b;
        b.x = hi ? qz : qx;
        b.y = hi ? y2 : qy;

        // D = A*B + C = x^2 + y^2 - 2 x.q  (full squared distance tile)
        v8f c = __builtin_amdgcn_wmma_f32_16x16x4_f32(
                /*neg_a=*/false, a, /*neg_b=*/false, b,
                /*c_mod=*/(short)0, x2v, /*reuse_a=*/false, /*reuse_b=*/false);

#pragma unroll
        for (int r = 0; r < 8; ++r) {
            const int ci = __float_as_int(c[r]);
            rmin[r] = (ci < rmin[r]) ? ci : rmin[r];    // v_min_i32 / v_min3_i32
        }
    }

    // Min across the 16 lanes of each half (columns of the tile sweep),
    // still in integer domain; clamp to 0 once at the end.
    float sum = 0.0f;
#pragma unroll
    for (int r = 0; r < 8; ++r) {
        int v = rmin[r];
        int t;
        t = __shfl_xor(v, 1, 32);  v = (t < v) ? t : v;
        t = __shfl_xor(v, 2, 32);  v = (t < v) ? t : v;
        t = __shfl_xor(v, 4, 32);  v = (t < v) ? t : v;
        t = __shfl_xor(v, 8, 32);  v = (t < v) ? t : v;
        sum += fmaxf(__int_as_float(v), 0.0f);
    }

    // Combine halves -> sum of all 16 row-mins of this tile.
    sum += __shfl_xor(sum, 16, 32);

    if (lane == 0) atomicAdd(accum + bs, sum);
}

// out[s*B + b] = (sum_min_fwd[b,s] + sum_min_bwd[b,s]) / NPTS   (chamfer.T)
__global__ void chamfer_finalize_kernel(const float* __restrict__ ws,
                                        float* __restrict__ out) {
    int t = threadIdx.x;
    if (t < BDIM * SDIM) {
        int s = t / BDIM;
        int b = t % BDIM;
        int bs = b * SDIM + s;
        out[t] = (ws[bs] + ws[BDIM * SDIM + bs]) * (1.0f / (float)NPTS);
    }
}

extern "C" void kernel_launch(void* const* d_in, const int* in_sizes, int n_in,
                              void* d_out, int out_size, void* d_ws, size_t ws_size,
                              hipStream_t stream) {
    const float* outp = (const float*)d_in[0];   // [B,S,N,3]
    const float* tgtp = (const float*)d_in[1];   // [B,S,M,3]
    float* ws  = (float*)d_ws;                   // [2*B*S] accumulators
    float* out = (float*)d_out;                  // [S,B]

    chamfer_zero_kernel<<<1, 64, 0, stream>>>(ws);

    // B*S*(N/16) waves, 8 waves per 256-thread block -> 512 blocks.
    const int totalWaves = BDIM * SDIM * (NPTS / 16);
    const int blocks = totalWaves / 8;
    // dist1: P=output points, Q=target points
    chamfer_nn_kernel<<<blocks, 256, 0, stream>>>(outp, tgtp, ws);
    // dist2: roles swapped
    chamfer_nn_kernel<<<blocks, 256, 0, stream>>>(tgtp, outp, ws + BDIM * SDIM);

    chamfer_finalize_kernel<<<1, 32, 0, stream>>>(ws, out);
}